// EncoderSAE_74741020885578
// MI455X (gfx1250) — compile-verified
//
#include <hip/hip_runtime.h>
#include <hip/hip_bf16.h>

#define SAE_B    2048
#define SAE_I    1024
#define SAE_D    32768
#define SAE_TOPK 64

// GEMM tiling
#define BM  128
#define BN  64
#define KC  16
#define LDP 20   // LDS row pitch (floats): 16B-aligned rows AND conflict-free (20*m mod 64 distinct)

typedef __attribute__((ext_vector_type(2))) float v2f;
typedef __attribute__((ext_vector_type(8))) float v8f;
typedef int sae_v4i __attribute__((vector_size(16)));   // matches builtin param pointee

#if __has_builtin(__builtin_amdgcn_global_load_async_to_lds_b128) && \
    __has_builtin(__builtin_amdgcn_s_wait_asynccnt)
#define SAE_ASYNC 1
#else
#define SAE_ASYNC 0
#endif

// Stage 16B global -> LDS. Async (ASYNCcnt) on CDNA5 when the builtin exists.
__device__ __forceinline__ void stage_b128(const float* gsrc, float* ldst) {
#if SAE_ASYNC
    __builtin_amdgcn_global_load_async_to_lds_b128(
        (__attribute__((address_space(1))) sae_v4i*)(sae_v4i*)const_cast<float*>(gsrc),
        (__attribute__((address_space(3))) sae_v4i*)(sae_v4i*)ldst,
        0, 0);
#else
    const float4 v = *reinterpret_cast<const float4*>(gsrc);
    ldst[0] = v.x; ldst[1] = v.y; ldst[2] = v.z; ldst[3] = v.w;
#endif
}

__device__ __forceinline__ void wait_stage_3() {
#if SAE_ASYNC
    __builtin_amdgcn_s_wait_asynccnt(3);
#endif
}
__device__ __forceinline__ void wait_stage_0() {
#if SAE_ASYNC
    __builtin_amdgcn_s_wait_asynccnt(0);
#endif
}

// -----------------------------------------------------------------------------
// C[M x N] = act( A[M x K] @ W[N x K]^T ), fp32 WMMA 16x16x4.
// Block tile 128(M) x 64(N), 256 thr = 8 waves as 4(M) x 2(N);
// each wave computes 32x32 (2x2 accumulators) -> 2x fragment reuse.
// Double-buffered LDS, async global->LDS staging overlapped with wmma.
// -----------------------------------------------------------------------------
__global__ void __launch_bounds__(256) sae_wmma_gemm(
    const float* __restrict__ A,
    const float* __restrict__ W,
    float* __restrict__ C,
    int K, int N, int do_relu)
{
    __shared__ float As[2][BM][LDP];
    __shared__ float Ws[2][BN][LDP];

    const int tid  = threadIdx.x;
    const int m0   = blockIdx.y * BM;
    const int n0   = blockIdx.x * BN;
    const int wave = tid >> 5;
    const int lane = tid & 31;
    const int l16  = lane & 15;
    const int lh   = lane >> 4;       // half-wave 0/1
    const int wm   = wave & 3;        // wave M sub-tile: rows wm*32..+31
    const int wn   = wave >> 2;       // wave N sub-tile: cols wn*32..+31

    // cooperative staging: 256 thr x b128; A needs 2 issues (128x16), W one (64x16)
    const int lr = tid >> 2;          // 0..63
    const int lc = (tid & 3) << 2;    // 0,4,8,12

    const float* Ag = A + (size_t)(m0 + lr) * K + lc;
    const float* Wg = W + (size_t)(n0 + lr) * K + lc;

    v8f acc[2][2] = {};
    const int NC = K / KC;

    // prologue: stage chunk 0 into buffer 0
    stage_b128(Ag,                   &As[0][lr][lc]);
    stage_b128(Ag + (size_t)64 * K,  &As[0][lr + 64][lc]);
    stage_b128(Wg,                   &Ws[0][lr][lc]);

    for (int c = 0; c < NC; ++c) {
        const int cur = c & 1;
        if (c > 0) __syncthreads();          // all reads of buf[cur^1] done
        if (c + 1 < NC) {                    // stage next chunk into other buffer
            const int ko = (c + 1) * KC;
            stage_b128(Ag + ko,                  &As[cur ^ 1][lr][lc]);
            stage_b128(Ag + (size_t)64 * K + ko, &As[cur ^ 1][lr + 64][lc]);
            stage_b128(Wg + ko,                  &Ws[cur ^ 1][lr][lc]);
            wait_stage_3();                  // cur's 3 ops done; next's 3 in flight
        } else {
            wait_stage_0();
        }
        __syncthreads();                     // cur tile visible to all waves

        // batch-load all fragments for this chunk, then 16 back-to-back wmmas
        v2f af[2][4], bf[2][4];
        #pragma unroll
        for (int kk = 0; kk < 4; ++kk) {
            const int kb = kk * 4 + 2 * lh;  // ISA 7.12.2 fp32 A/B layout
            #pragma unroll
            for (int t = 0; t < 2; ++t) {
                af[t][kk].x = As[cur][wm * 32 + t * 16 + l16][kb];
                af[t][kk].y = As[cur][wm * 32 + t * 16 + l16][kb + 1];
                bf[t][kk].x = Ws[cur][wn * 32 + t * 16 + l16][kb];
                bf[t][kk].y = Ws[cur][wn * 32 + t * 16 + l16][kb + 1];
            }
        }
        #pragma unroll
        for (int kk = 0; kk < 4; ++kk)
            #pragma unroll
            for (int ti = 0; ti < 2; ++ti)
                #pragma unroll
                for (int tj = 0; tj < 2; ++tj)
                    acc[ti][tj] = __builtin_amdgcn_wmma_f32_16x16x4_f32(
                        false, af[ti][kk], false, bf[tj][kk],
                        (short)0, acc[ti][tj], false, false);
    }

    // C/D layout: VGPR g -> row g (lanes 0-15) / g+8 (lanes 16-31), col = l16
    #pragma unroll
    for (int ti = 0; ti < 2; ++ti)
        #pragma unroll
        for (int tj = 0; tj < 2; ++tj)
            #pragma unroll
            for (int g = 0; g < 8; ++g) {
                const int m = m0 + wm * 32 + ti * 16 + g + 8 * lh;
                const int n = n0 + wn * 32 + tj * 16 + l16;
                float v = acc[ti][tj][g];
                if (do_relu) v = fmaxf(v, 0.0f);
                C[(size_t)m * N + n] = v;
            }
}

// -----------------------------------------------------------------------------
// Exact per-row top-k threshold: 4-pass radix select on float bits
// (values >= 0 after relu => unsigned-monotonic). One 256-thr block per row.
// -----------------------------------------------------------------------------
__global__ void __launch_bounds__(256) sae_topk_thresh(
    const float* __restrict__ raw, float* __restrict__ thresh)
{
    __shared__ unsigned hist[256];
    __shared__ unsigned s_sel, s_rem;

    const int b = blockIdx.x;
    const unsigned tid = threadIdx.x;
    const float4* row4 = reinterpret_cast<const float4*>(raw + (size_t)b * SAE_D);

    unsigned prefix = 0u, mask = 0u, rem = SAE_TOPK;

    for (int pass = 0; pass < 4; ++pass) {
        const int shift = 24 - 8 * pass;
        hist[tid] = 0u;
        __syncthreads();
        for (int i = tid; i < SAE_D / 4; i += 256) {
            const float4 v = row4[i];
            const unsigned u0 = __float_as_uint(v.x);
            const unsigned u1 = __float_as_uint(v.y);
            const unsigned u2 = __float_as_uint(v.z);
            const unsigned u3 = __float_as_uint(v.w);
            if ((u0 & mask) == prefix) atomicAdd(&hist[(u0 >> shift) & 0xFFu], 1u);
            if ((u1 & mask) == prefix) atomicAdd(&hist[(u1 >> shift) & 0xFFu], 1u);
            if ((u2 & mask) == prefix) atomicAdd(&hist[(u2 >> shift) & 0xFFu], 1u);
            if ((u3 & mask) == prefix) atomicAdd(&hist[(u3 >> shift) & 0xFFu], 1u);
        }
        __syncthreads();
        if (tid == 0) {
            unsigned cum = 0u, sel = 0u;
            for (int bkt = 255; bkt >= 0; --bkt) {
                const unsigned c = hist[bkt];
                if (cum + c >= rem) { sel = (unsigned)bkt; break; }
                cum += c;
            }
            s_sel = sel;
            s_rem = rem - cum;
        }
        __syncthreads();
        prefix |= s_sel << shift;
        mask   |= 0xFFu << shift;
        rem     = s_rem;
        __syncthreads();
    }
    if (tid == 0) thresh[b] = __uint_as_float(prefix);   // exact k-th largest
}

__global__ void zero_l0(float* __restrict__ p)
{
    if (threadIdx.x == 0 && blockIdx.x == 0) *p = 0.0f;
}

// -----------------------------------------------------------------------------
// features = (raw >= thresh[b]) ? raw : 0 ; l0 += count(features>0)/B
// -----------------------------------------------------------------------------
__global__ void __launch_bounds__(256) sae_scatter(
    const float* __restrict__ raw,
    const float* __restrict__ thresh,
    float* __restrict__ feat,
    float* __restrict__ l0)
{
    const float4* raw4 = reinterpret_cast<const float4*>(raw);
    float4* feat4 = reinterpret_cast<float4*>(feat);
    const size_t total4 = (size_t)SAE_B * SAE_D / 4;
    const size_t stride = (size_t)gridDim.x * blockDim.x;
    unsigned cnt = 0;
    for (size_t i = (size_t)blockIdx.x * blockDim.x + threadIdx.x;
         i < total4; i += stride) {
        const int   b = (int)(i / (SAE_D / 4));
        const float t = thresh[b];
        const float4 v = raw4[i];
        float4 f;
        f.x = (v.x >= t) ? v.x : 0.0f;
        f.y = (v.y >= t) ? v.y : 0.0f;
        f.z = (v.z >= t) ? v.z : 0.0f;
        f.w = (v.w >= t) ? v.w : 0.0f;
        feat4[i] = f;
        cnt += (f.x > 0.0f) + (f.y > 0.0f) + (f.z > 0.0f) + (f.w > 0.0f);
    }
    __shared__ unsigned sred[256];
    sred[threadIdx.x] = cnt;
    __syncthreads();
    for (int off = 128; off > 0; off >>= 1) {
        if ((int)threadIdx.x < off) sred[threadIdx.x] += sred[threadIdx.x + off];
        __syncthreads();
    }
    if (threadIdx.x == 0 && sred[0] != 0u)
        atomicAdd(l0, (float)sred[0] * (1.0f / (float)SAE_B));
}

// -----------------------------------------------------------------------------
// d_out layout (floats): recon[B*I] | features[B*D] | l0[1] | thresh[B] | raw[B*D]
// -----------------------------------------------------------------------------
extern "C" void kernel_launch(void* const* d_in, const int* in_sizes, int n_in,
                              void* d_out, int out_size, void* d_ws, size_t ws_size,
                              hipStream_t stream) {
    (void)in_sizes; (void)n_in; (void)out_size; (void)d_ws; (void)ws_size;

    const float* x     = (const float*)d_in[0];   // [B, I]
    const float* W_enc = (const float*)d_in[1];   // [D, I]
    const float* W_dec = (const float*)d_in[2];   // [I, D]

    float* out    = (float*)d_out;
    float* recon  = out;
    float* feat   = recon + (size_t)SAE_B * SAE_I;
    float* l0     = feat  + (size_t)SAE_B * SAE_D;
    float* thresh = l0 + 1;
    float* raw    = thresh + SAE_B;

    const dim3 blk(256);

    // 1. encoder: raw = relu(x @ W_enc^T)  [2048 x 32768], K = 1024
    {
        const dim3 grid(SAE_D / BN, SAE_B / BM);
        sae_wmma_gemm<<<grid, blk, 0, stream>>>(x, W_enc, raw, SAE_I, SAE_D, 1);
    }
    // 2. exact per-row top-64 threshold
    sae_topk_thresh<<<SAE_B, blk, 0, stream>>>(raw, thresh);
    // 3. l0 init + feature scatter
    zero_l0<<<1, 1, 0, stream>>>(l0);
    sae_scatter<<<1024, blk, 0, stream>>>(raw, thresh, feat, l0);
    // 4. decoder: recon = features @ W_dec^T  [2048 x 1024], K = 32768
    {
        const dim3 grid(SAE_I / BN, SAE_B / BM);
        sae_wmma_gemm<<<grid, blk, 0, stream>>>(feat, W_dec, recon, SAE_D, SAE_I, 0);
    }
}